// SPrompts_84095459655779
// MI455X (gfx1250) — compile-verified
//
#include <hip/hip_runtime.h>
#include <stdint.h>

typedef float v2f __attribute__((ext_vector_type(2)));
typedef float v8f __attribute__((ext_vector_type(8)));

#define Bq     512
#define Lq     12
#define Dq     768
#define Tq     20
#define NKq    100
#define NPq    10
#define NKEYS  2000          // T*NK
#define MT     64            // query rows per block (4 waves x 16)
#define NC     80            // keys per chunk (5 tiles of 16 per wave)
#define KC     32            // D per LDS chunk
#define LDP    36            // padded LDS row stride (floats): mult of 4, 16 distinct banks
#define NCHUNK (Dq / KC)     // 24
#define NSPLIT 5             // key splits per (l, mtile)
#define KEYS_PER_SPLIT 400   // NKEYS / NSPLIT
#define CHUNKS_PER_SPLIT 5   // KEYS_PER_SPLIT / NC
#define AV4    ((MT * (KC / 4)) / 128)       // 4 float4 per thread for A tile
#define KV4    ((NC * (KC / 4)) / 128)       // 5 float4 per thread for keys tile
// per-wave async ops in flight per chunk = AV4 + KV4 = 9

// ---------------------------------------------------------------- async copy helpers
__device__ __forceinline__ void sp_async_b128(const float* gp, float* lp) {
    // vdst = LDS byte address (low 32 bits of the flat shared pointer),
    // vaddr = 64-bit global address, GV mode (no SADDR).
    unsigned           loff = (unsigned)(unsigned long long)(const void*)lp;
    unsigned long long gaddr = (unsigned long long)(const void*)gp;
    asm volatile("global_load_async_to_lds_b128 %0, %1, off"
                 :: "v"(loff), "v"(gaddr) : "memory");
}

// ---------------------------------------------------------------- init
__global__ void SP_init_kernel(unsigned long long* __restrict__ best,
                               float* __restrict__ out_scalar) {
    int i = blockIdx.x * blockDim.x + threadIdx.x;
    if (i < Lq * Bq) best[i] = 0ull;
    if (i == 0) *out_scalar = 0.0f;
}

// ---------------------------------------------------------------- 0.5*||key||^2
__global__ __launch_bounds__(256) void SP_k2_kernel(const float* __restrict__ keys,
                                                    float* __restrict__ k2h) {
    int row  = blockIdx.x * 8 + (threadIdx.x >> 5);   // one wave32 per key row
    int lane = threadIdx.x & 31;
    if (row >= Lq * NKEYS) return;
    const float4* p = (const float4*)(keys + (size_t)row * Dq);
    float s = 0.0f;
#pragma unroll
    for (int i = 0; i < Dq / 128; ++i) {              // 6 float4 per lane
        float4 v = p[lane + i * 32];
        s += v.x * v.x + v.y * v.y + v.z * v.z + v.w * v.w;
    }
#pragma unroll
    for (int m = 16; m >= 1; m >>= 1) s += __shfl_xor(s, m, 32);
    if (lane == 0) k2h[row] = 0.5f * s;
}

// ---------------------------------------------------------------- chunk loader (async)
__device__ __forceinline__ void sp_issue_chunk(const float* __restrict__ gA,
                                               const float* __restrict__ gK,
                                               float* __restrict__ sA,
                                               float* __restrict__ sK, int tid) {
#pragma unroll
    for (int j = 0; j < AV4; ++j) {
        int id = tid + j * 128;
        int row = id >> 3, c4 = (id & 7) << 2;        // KC/4 = 8 vec4 per row
        sp_async_b128(gA + (size_t)row * (Lq * Dq) + c4, sA + row * LDP + c4);
    }
#pragma unroll
    for (int j = 0; j < KV4; ++j) {
        int id = tid + j * 128;
        int row = id >> 3, c4 = (id & 7) << 2;
        sp_async_b128(gK + (size_t)row * Dq + c4, sK + row * LDP + c4);
    }
}

// ---------------------------------------------------------------- WMMA compute on one chunk
__device__ __forceinline__ void sp_compute_chunk(const float* __restrict__ sA,
                                                 const float* __restrict__ sK,
                                                 int wave, int ln, int half, v8f acc[5]) {
    const int arow = (wave * 16 + ln) * LDP + 2 * half;
    const int brow = ln * LDP + 2 * half;
#pragma unroll
    for (int kk = 0; kk < KC; kk += 4) {
        // A 16x4 f32 fragment: lane(m=ln) holds K = kk + 2*half .. +1
        v2f a = *(const v2f*)&sA[arow + kk];
#pragma unroll
        for (int nt = 0; nt < 5; ++nt) {
            // B 4x16 f32 fragment: lane(n=ln) holds K = kk + 2*half .. +1
            v2f b = *(const v2f*)&sK[nt * 16 * LDP + brow + kk];
            acc[nt] = __builtin_amdgcn_wmma_f32_16x16x4_f32(
                false, a, false, b, (short)0, acc[nt], false, false);
        }
    }
}

// ---------------------------------------------------------------- fused GEMM + argmax
__global__ __launch_bounds__(128) void SP_gemm_argmax_kernel(
    const float* __restrict__ xq, const float* __restrict__ keys,
    const float* __restrict__ k2h, unsigned long long* __restrict__ best) {

    __shared__ float lds_a[2][MT * LDP];   // 2 x 9.2 KB
    __shared__ float lds_k[2][NC * LDP];   // 2 x 11.5 KB

    const int bid   = blockIdx.x;
    const int l     = bid / (8 * NSPLIT);
    const int rem   = bid % (8 * NSPLIT);
    const int b0    = (rem / NSPLIT) * MT;
    const int nbase = (rem % NSPLIT) * KEYS_PER_SPLIT;

    const int tid  = threadIdx.x;
    const int lane = tid & 31;
    const int wave = tid >> 5;
    const int half = lane >> 4;
    const int ln   = lane & 15;

    float bestv[8];
    int   bestidx[8];
#pragma unroll
    for (int r = 0; r < 8; ++r) { bestv[r] = -3.4e38f; bestidx[r] = 0; }

    for (int ci = 0; ci < CHUNKS_PER_SPLIT; ++ci) {
        const int n0 = nbase + ci * NC;
        v8f acc[5] = {};

        const float* gA = xq + ((size_t)b0 * Lq + l) * Dq;           // row stride L*D
        const float* gK = keys + ((size_t)l * NKEYS + n0) * Dq;      // row stride D

        // prologue: fill buffer 0 for chunk 0
        sp_issue_chunk(gA, gK, lds_a[0], lds_k[0], tid);

        for (int kc = 0; kc < NCHUNK; ++kc) {
            const int cur = kc & 1;
            if (kc + 1 < NCHUNK) {
                // overlap next chunk's global->LDS DMA with this chunk's WMMA
                sp_issue_chunk(gA + (kc + 1) * KC, gK + (kc + 1) * KC,
                               lds_a[cur ^ 1], lds_k[cur ^ 1], tid);
                // async ops complete in order: <=9 outstanding => current chunk landed
                asm volatile("s_wait_asynccnt 0x9" ::: "memory");
            } else {
                asm volatile("s_wait_asynccnt 0x0" ::: "memory");
            }
            __syncthreads();
            sp_compute_chunk(lds_a[cur], lds_k[cur], wave, ln, half, acc);
            __syncthreads();   // all waves done reading buf[cur] before it is refilled
        }

        // fuse score = dot - 0.5*||k||^2 and running argmax.
        // C layout: VGPR r, lane -> (M = r + 8*half, N = ln); N fixed per lane/tile.
#pragma unroll
        for (int nt = 0; nt < 5; ++nt) {
            int   gk = n0 + nt * 16 + ln;
            float h  = k2h[l * NKEYS + gk];
#pragma unroll
            for (int r = 0; r < 8; ++r) {
                float v = acc[nt][r] - h;
                if (v > bestv[r] || (v == bestv[r] && gk < bestidx[r])) {
                    bestv[r] = v; bestidx[r] = gk;
                }
            }
        }
    }

    // reduce across the 16 lanes of each half (xor masks < 16 stay within a half)
#pragma unroll
    for (int r = 0; r < 8; ++r) {
        float v = bestv[r];
        int   i = bestidx[r];
#pragma unroll
        for (int m = 1; m < 16; m <<= 1) {
            float ov = __shfl_xor(v, m, 32);
            int   oi = __shfl_xor(i, m, 32);
            if (ov > v || (ov == v && oi < i)) { v = ov; i = oi; }
        }
        if (ln == 0) {
            int row = b0 + wave * 16 + r + 8 * half;
            unsigned int key = __float_as_uint(v);
            key = (key & 0x80000000u) ? ~key : (key | 0x80000000u);  // order-preserving map
            unsigned long long packed =
                ((unsigned long long)key << 32) | (unsigned int)(~(unsigned int)i);
            atomicMax(&best[(size_t)l * Bq + row], packed);          // ties -> smallest idx
        }
    }
}

// ---------------------------------------------------------------- gather prompts
__global__ __launch_bounds__(256) void SP_gather_kernel(
    const unsigned long long* __restrict__ best,
    const float* __restrict__ prompts, float* __restrict__ out) {
    int i = blockIdx.x;               // i = l*B + b
    int l = i / Bq;
    unsigned int idx = ~(unsigned int)best[i];
    int task = (int)(idx / NKq);
    if (task >= Tq) task = Tq - 1;
    const float4* src = (const float4*)(prompts + ((size_t)(l * Tq + task) * NPq) * Dq);
    float4*       dst = (float4*)(out + (size_t)i * NPq * Dq);
#pragma unroll 2
    for (int j = threadIdx.x; j < (NPq * Dq) / 4; j += 256) dst[j] = src[j];
}

// ---------------------------------------------------------------- launch
extern "C" void kernel_launch(void* const* d_in, const int* in_sizes, int n_in,
                              void* d_out, int out_size, void* d_ws, size_t ws_size,
                              hipStream_t stream) {
    const float* xq      = (const float*)d_in[0];
    const float* keys    = (const float*)d_in[1];
    const float* prompts = (const float*)d_in[2];
    float*       out     = (float*)d_out;

    float* k2h = (float*)d_ws;                                   // 24000 floats
    unsigned long long* best =
        (unsigned long long*)((char*)d_ws + (size_t)Lq * NKEYS * sizeof(float));  // 6144 u64

    SP_init_kernel<<<(Lq * Bq + 255) / 256, 256, 0, stream>>>(best, out + (out_size - 1));
    SP_k2_kernel<<<(Lq * NKEYS) / 8, 256, 0, stream>>>(keys, k2h);
    SP_gemm_argmax_kernel<<<Lq * 8 * NSPLIT, 128, 0, stream>>>(xq, keys, k2h, best);
    SP_gather_kernel<<<Lq * Bq, 256, 0, stream>>>(best, prompts, out);
}